// GCNBMPEncoder_63264868270126
// MI455X (gfx1250) — compile-verified
//
#include <hip/hip_runtime.h>
#include <hip/hip_bf16.h>
#include <math.h>

// ---------------------------------------------------------------------------
// RGCN (3 layers, DIM=64, 4 relations) + highway gating for gfx1250 (CDNA5).
//   Phase 1 per layer : wave-per-edge gather + f32 atomic scatter into
//                       (dst, rel) buckets  (memory/atomic bound, ~1GB/layer)
//   Phase 2 per layer : node update GEMM  [N,320] x [320,64] via
//                       V_WMMA_F32_16X16X4_F32, degree-normalization fused
//                       into the A-fragment load, sigmoid epilogue
//   Phase 3 per layer : highway GEMMs [N,128] x [128,64] (proj + gate) via
//                       WMMA, relu/sigmoid/blend fused epilogue
// B-matrix weights live transposed in LDS with a padded stride so each lane's
// {W[kk][col], W[kk+1][col]} pair is one aligned, bank-conflict-free
// ds_load_b64 straight into an even VGPR pair (no repack moves).
// ---------------------------------------------------------------------------

typedef __attribute__((ext_vector_type(2))) float v2f;
typedef __attribute__((ext_vector_type(8))) float v8f;

#define DIM   64
#define NREL  4
#define KCONV 320        // 256 (upd) + 64 (self)
#define KHW   128        // 64 (hidden) + 64 (prev)
#define STRC  324        // padded K-stride for conv weights   (324 % 64 == 4, even)
#define STRH  132        // padded K-stride for highway weights(132 % 64 == 4, even)

__device__ __forceinline__ float sigmoidf_(float v) {
    return 1.0f / (1.0f + __expf(-v));
}

// ---------------------------------------------------------------------------
// Edge scatter: one wave32 per edge. Each lane carries 2 floats of the 64-wide
// message row (float2 per lane = full 256B row per wave), atomically
// accumulated into agg[(dst*4+rel)*64 + :]. Lane 0 accumulates the degree.
// ---------------------------------------------------------------------------
__global__ void rgcn_scatter_kernel(const float* __restrict__ x,
                                    const float* __restrict__ ew,
                                    const int*   __restrict__ src,
                                    const int*   __restrict__ dst,
                                    const int*   __restrict__ rel,
                                    float*       __restrict__ agg,
                                    float*       __restrict__ deg,
                                    int nEdges)
{
    int tid  = blockIdx.x * blockDim.x + threadIdx.x;
    int e    = tid >> 5;
    int lane = tid & 31;
    if (e >= nEdges) return;
    int   s = src[e];
    int   d = dst[e];
    int   r = rel[e];
    float w = ew[e];
    float2 v = ((const float2*)(x + (size_t)s * DIM))[lane];
    float* aggRow = agg + ((size_t)d * NREL + r) * DIM;
    atomicAdd(aggRow + 2 * lane,     v.x * w);
    atomicAdd(aggRow + 2 * lane + 1, v.y * w);
    if (lane == 0) atomicAdd(deg + (size_t)d * NREL + r, w);
}

// ---------------------------------------------------------------------------
// Node update: hidden = sigmoid( upd @ lin_w + x @ self_w + lin_b + self_b )
// upd = agg / (deg + eps), fused into A-fragment load.
// One wave handles 16 node rows x 64 output cols. K = 320 in steps of 4,
// split into a straight-line agg phase (k<256) and x phase (k>=256).
// ---------------------------------------------------------------------------
__global__ void rgcn_update_kernel(const float* __restrict__ agg,
                                   const float* __restrict__ deg,
                                   const float* __restrict__ x,
                                   const float* __restrict__ lin_w,   // [256,64]
                                   const float* __restrict__ lin_b,   // [64]
                                   const float* __restrict__ self_w,  // [64,64]
                                   const float* __restrict__ self_b,  // [64]
                                   float*       __restrict__ hidden,
                                   int nNodes)
{
    extern __shared__ float smem[];
    float* WT = smem;                  // transposed weights [64][STRC]
    float* RD = smem + DIM * STRC;     // [8 waves][64] reciprocal degrees

    const int t = threadIdx.x;
    // stage weights transposed: WT[c][k] = W[k][c]; global reads coalesced
    for (int i = t; i < KCONV * DIM; i += blockDim.x) {
        int k = i >> 6;
        int c = i & 63;
        float v = (k < NREL * DIM) ? lin_w[i] : self_w[i - NREL * DIM * DIM];
        WT[(size_t)c * STRC + k] = v;
    }

    const int wave = t >> 5;
    const int lane = t & 31;
    const int gw   = blockIdx.x * (blockDim.x >> 5) + wave;
    const long base = (long)gw * 16;

    if (base < nNodes) {   // wave-uniform predicate; N is a multiple of 16
        // deg is node-major: this wave's 64 values are contiguous
        float2 dv = ((const float2*)(deg + base * NREL))[lane];
        RD[wave * 64 + 2 * lane]     = 1.0f / (dv.x + 1e-10f);
        RD[wave * 64 + 2 * lane + 1] = 1.0f / (dv.y + 1e-10f);
    }
    __syncthreads();
    if (base >= nNodes) return;

    const int m     = lane & 15;     // node-local row / column within tile
    const int khalf = lane >> 4;     // 0 -> K,K+1 ; 1 -> K+2,K+3
    const float* aggRow = agg + (base + m) * (size_t)(NREL * DIM);
    const float* xRow   = x   + (base + m) * (size_t)DIM;
    const float* B0 = WT + (size_t)(m)      * STRC;
    const float* B1 = WT + (size_t)(16 + m) * STRC;
    const float* B2 = WT + (size_t)(32 + m) * STRC;
    const float* B3 = WT + (size_t)(48 + m) * STRC;
    const float* rd = RD + wave * 64 + m * NREL;

    v8f a0 = {}, a1 = {}, a2 = {}, a3 = {};
    // ---- phase 1: normalized aggregate, K = 0..255 ----
    for (int k = 0; k < NREL * DIM; k += 4) {
        int kk = k + 2 * khalf;
        float  sc = rd[k >> 6];
        float2 av = *(const float2*)(aggRow + kk);
        v2f a; a.x = av.x * sc; a.y = av.y * sc;
        v2f b0 = *(const v2f*)(B0 + kk);
        v2f b1 = *(const v2f*)(B1 + kk);
        v2f b2 = *(const v2f*)(B2 + kk);
        v2f b3 = *(const v2f*)(B3 + kk);
        a0 = __builtin_amdgcn_wmma_f32_16x16x4_f32(false, a, false, b0, (short)0, a0, false, false);
        a1 = __builtin_amdgcn_wmma_f32_16x16x4_f32(false, a, false, b1, (short)0, a1, false, false);
        a2 = __builtin_amdgcn_wmma_f32_16x16x4_f32(false, a, false, b2, (short)0, a2, false, false);
        a3 = __builtin_amdgcn_wmma_f32_16x16x4_f32(false, a, false, b3, (short)0, a3, false, false);
    }
    // ---- phase 2: self connection, K = 256..319 ----
    for (int k = 0; k < DIM; k += 4) {
        int kk = k + 2 * khalf;
        float2 av = *(const float2*)(xRow + kk);
        v2f a; a.x = av.x; a.y = av.y;
        int kw = NREL * DIM + kk;
        v2f b0 = *(const v2f*)(B0 + kw);
        v2f b1 = *(const v2f*)(B1 + kw);
        v2f b2 = *(const v2f*)(B2 + kw);
        v2f b3 = *(const v2f*)(B3 + kw);
        a0 = __builtin_amdgcn_wmma_f32_16x16x4_f32(false, a, false, b0, (short)0, a0, false, false);
        a1 = __builtin_amdgcn_wmma_f32_16x16x4_f32(false, a, false, b1, (short)0, a1, false, false);
        a2 = __builtin_amdgcn_wmma_f32_16x16x4_f32(false, a, false, b2, (short)0, a2, false, false);
        a3 = __builtin_amdgcn_wmma_f32_16x16x4_f32(false, a, false, b3, (short)0, a3, false, false);
    }

#define CONV_EPILOGUE(ACC, TILE)                                           \
    {                                                                      \
        int   col = (TILE) * 16 + m;                                       \
        float bia = lin_b[col] + self_b[col];                              \
        _Pragma("unroll")                                                  \
        for (int r = 0; r < 8; ++r) {                                      \
            long  row = base + khalf * 8 + r;                              \
            hidden[row * DIM + col] = sigmoidf_(ACC[r] + bia);             \
        }                                                                  \
    }
    CONV_EPILOGUE(a0, 0) CONV_EPILOGUE(a1, 1) CONV_EPILOGUE(a2, 2) CONV_EPILOGUE(a3, 3)
#undef CONV_EPILOGUE
}

// ---------------------------------------------------------------------------
// Highway: cat = [hidden, prev] (K=128)
//   proj = relu(cat @ pw + pb); gate = sigmoid(cat @ tw + tb)
//   out  = gate*proj + (1-gate)*hidden
// ---------------------------------------------------------------------------
__global__ void highway_kernel(const float* __restrict__ hid,
                               const float* __restrict__ prev,
                               const float* __restrict__ pw,  // [128,64]
                               const float* __restrict__ pb,  // [64]
                               const float* __restrict__ tw,  // [128,64]
                               const float* __restrict__ tb,  // [64]
                               float*       __restrict__ outp,
                               int nNodes)
{
    extern __shared__ float smem[];
    float* WP = smem;               // transposed [64][STRH]
    float* WTT = smem + DIM * STRH; // transposed [64][STRH]

    const int t = threadIdx.x;
    for (int i = t; i < KHW * DIM; i += blockDim.x) {
        int k = i >> 6;
        int c = i & 63;
        WP [(size_t)c * STRH + k] = pw[i];
        WTT[(size_t)c * STRH + k] = tw[i];
    }
    __syncthreads();

    const int wave = t >> 5;
    const int lane = t & 31;
    const int gw   = blockIdx.x * (blockDim.x >> 5) + wave;
    const long base = (long)gw * 16;
    if (base >= nNodes) return;

    const int m     = lane & 15;
    const int khalf = lane >> 4;
    const float* hRow = hid  + (base + m) * (size_t)DIM;
    const float* pRow = prev + (base + m) * (size_t)DIM;
    const float* P0 = WP  + (size_t)(m)      * STRH;
    const float* P1 = WP  + (size_t)(16 + m) * STRH;
    const float* P2 = WP  + (size_t)(32 + m) * STRH;
    const float* P3 = WP  + (size_t)(48 + m) * STRH;
    const float* T0 = WTT + (size_t)(m)      * STRH;
    const float* T1 = WTT + (size_t)(16 + m) * STRH;
    const float* T2 = WTT + (size_t)(32 + m) * STRH;
    const float* T3 = WTT + (size_t)(48 + m) * STRH;

    v8f p0 = {}, p1 = {}, p2 = {}, p3 = {};
    v8f g0 = {}, g1 = {}, g2 = {}, g3 = {};

#define HW_STEP(AROW, KOFF)                                                              \
    for (int k = 0; k < DIM; k += 4) {                                                   \
        int kk = k + 2 * khalf;                                                          \
        float2 av = *(const float2*)((AROW) + kk);                                       \
        v2f a; a.x = av.x; a.y = av.y;                                                   \
        int kw = kk + (KOFF);                                                            \
        v2f bp0 = *(const v2f*)(P0 + kw);                                                \
        v2f bp1 = *(const v2f*)(P1 + kw);                                                \
        v2f bp2 = *(const v2f*)(P2 + kw);                                                \
        v2f bp3 = *(const v2f*)(P3 + kw);                                                \
        v2f bt0 = *(const v2f*)(T0 + kw);                                                \
        v2f bt1 = *(const v2f*)(T1 + kw);                                                \
        v2f bt2 = *(const v2f*)(T2 + kw);                                                \
        v2f bt3 = *(const v2f*)(T3 + kw);                                                \
        p0 = __builtin_amdgcn_wmma_f32_16x16x4_f32(false, a, false, bp0, (short)0, p0, false, false); \
        p1 = __builtin_amdgcn_wmma_f32_16x16x4_f32(false, a, false, bp1, (short)0, p1, false, false); \
        p2 = __builtin_amdgcn_wmma_f32_16x16x4_f32(false, a, false, bp2, (short)0, p2, false, false); \
        p3 = __builtin_amdgcn_wmma_f32_16x16x4_f32(false, a, false, bp3, (short)0, p3, false, false); \
        g0 = __builtin_amdgcn_wmma_f32_16x16x4_f32(false, a, false, bt0, (short)0, g0, false, false); \
        g1 = __builtin_amdgcn_wmma_f32_16x16x4_f32(false, a, false, bt1, (short)0, g1, false, false); \
        g2 = __builtin_amdgcn_wmma_f32_16x16x4_f32(false, a, false, bt2, (short)0, g2, false, false); \
        g3 = __builtin_amdgcn_wmma_f32_16x16x4_f32(false, a, false, bt3, (short)0, g3, false, false); \
    }
    HW_STEP(hRow, 0)      // first half of cat: this layer's hidden
    HW_STEP(pRow, DIM)    // second half of cat: previous gcn output
#undef HW_STEP

#define HW_EPILOGUE(PACC, GACC, TILE)                                      \
    {                                                                      \
        int   col = (TILE) * 16 + m;                                       \
        float pbv = pb[col];                                               \
        float tbv = tb[col];                                               \
        _Pragma("unroll")                                                  \
        for (int r = 0; r < 8; ++r) {                                      \
            long  row = base + khalf * 8 + r;                              \
            float pr = fmaxf(PACC[r] + pbv, 0.0f);                         \
            float gt = sigmoidf_(GACC[r] + tbv);                           \
            float h  = hid[row * DIM + col];                               \
            outp[row * DIM + col] = gt * pr + (1.0f - gt) * h;             \
        }                                                                  \
    }
    HW_EPILOGUE(p0, g0, 0) HW_EPILOGUE(p1, g1, 1)
    HW_EPILOGUE(p2, g2, 2) HW_EPILOGUE(p3, g3, 3)
#undef HW_EPILOGUE
}

// ---------------------------------------------------------------------------
extern "C" void kernel_launch(void* const* d_in, const int* in_sizes, int n_in,
                              void* d_out, int out_size, void* d_ws, size_t ws_size,
                              hipStream_t stream) {
    const float* node_feat   = (const float*)d_in[0];
    const float* edge_weight = (const float*)d_in[1];
    const float* conv_lin_w  = (const float*)d_in[2];   // [3,256,64]
    const float* conv_lin_b  = (const float*)d_in[3];   // [3,64]
    const float* conv_self_w = (const float*)d_in[4];   // [3,64,64]
    const float* conv_self_b = (const float*)d_in[5];   // [3,64]
    const float* hw_proj_w   = (const float*)d_in[6];   // [3,128,64]
    const float* hw_proj_b   = (const float*)d_in[7];   // [3,64]
    const float* hw_trans_w  = (const float*)d_in[8];   // [3,128,64]
    const float* hw_trans_b  = (const float*)d_in[9];   // [3,64]
    const int*   edge_src    = (const int*)d_in[10];
    const int*   edge_dst    = (const int*)d_in[11];
    const int*   edge_rel    = (const int*)d_in[12];
    float*       out         = (float*)d_out;

    const int N = in_sizes[0] / DIM;
    const int E = in_sizes[10];

    // workspace carve-up (floats)
    float* agg = (float*)d_ws;                            // N*4*64
    float* deg = agg + (size_t)N * NREL * DIM;            // N*4   (contiguous w/ agg)
    float* hA  = deg + (size_t)N * NREL;                  // N*64
    float* hB  = hA  + (size_t)N * DIM;                   // N*64
    float* cur = hB  + (size_t)N * DIM;                   // N*64

    const int scatterBlocks = (E + 7) / 8;                       // 8 edges / 256-thr block
    const int waves         = (N + 15) / 16;
    const int nodeBlocks    = (waves + 7) / 8;                   // 8 waves / block
    const size_t updLds = (size_t)(DIM * STRC + 8 * 64) * sizeof(float);  // 84992 B
    const size_t hwLds  = (size_t)(2 * DIM * STRH) * sizeof(float);       // 67584 B

    for (int i = 0; i < 3; ++i) {
        const float* xin  = (i == 0) ? node_feat : cur;
        const float* prev = (i == 0) ? node_feat : ((i == 1) ? hA : hB);
        float*       hid  = (i & 1) ? hB : hA;
        float*       gout = (i == 2) ? out : cur;

        hipMemsetAsync(agg, 0, (size_t)(N * NREL * DIM + N * NREL) * sizeof(float), stream);

        rgcn_scatter_kernel<<<scatterBlocks, 256, 0, stream>>>(
            xin, edge_weight, edge_src, edge_dst, edge_rel, agg, deg, E);

        rgcn_update_kernel<<<nodeBlocks, 256, updLds, stream>>>(
            agg, deg, xin,
            conv_lin_w  + (size_t)i * 256 * DIM, conv_lin_b  + (size_t)i * DIM,
            conv_self_w + (size_t)i * DIM * DIM, conv_self_b + (size_t)i * DIM,
            hid, N);

        highway_kernel<<<nodeBlocks, 256, hwLds, stream>>>(
            hid, prev,
            hw_proj_w  + (size_t)i * KHW * DIM, hw_proj_b  + (size_t)i * DIM,
            hw_trans_w + (size_t)i * KHW * DIM, hw_trans_b + (size_t)i * DIM,
            gout, N);
    }
}